// AttentionThresholdCoupler_81131932221977
// MI455X (gfx1250) — compile-verified
//
#include <hip/hip_runtime.h>
#include <hip/hip_bf16.h>
#include <math.h>

// ---------------------------------------------------------------------------
// AttentionThresholdCoupler, fused for MI455X (gfx1250, wave32, WMMA).
//
// Math collapse (softmax over a single key == 1.0):
//   ga = y1 + FFN1(y1),  y1 = LN1(g + A1)        A1 const (from coral)
//   ca = y2 + FFN2(y2),  y2 = LN2(M2*g + b2f)    M2=Wo2*Wv2, b2f folds coral
//   unified = tanh([ga|ca] @ int_w^T + int_b),   gw = cw = 1.0
//
// One wave per 16-row tile. LayerNorm is computed LANE-LOCALLY (each lane owns
// a full 8-float row) -- no shuffles / bpermutes. Residuals are folded into
// the 2nd FFN WMMA chain as an augmented [H | y] @ [W2^T ; I] product, so y
// is consumed straight from its A-operand registers. Only X2 and H need an
// LDS transpose (stride-20 pitch, conflict-free).
// ---------------------------------------------------------------------------

typedef __attribute__((ext_vector_type(2))) float v2f;
typedef __attribute__((ext_vector_type(4))) float v4f;
typedef __attribute__((ext_vector_type(8))) float v8f;

#define LDS_STRIDE 20   // floats per LDS row (16 cols + 4 pad)

__device__ __forceinline__ v8f wmma4(v2f a, v2f b, v8f c) {
  // D(16x16,f32) = A(16x4,f32) * B(4x16,f32) + C
  return __builtin_amdgcn_wmma_f32_16x16x4_f32(false, a, false, b, (short)0, c,
                                               false, false);
}

__device__ __forceinline__ v8f bcast8(float s) {
  v8f c;
#pragma unroll
  for (int i = 0; i < 8; ++i) c[i] = s;
  return c;
}

__device__ __forceinline__ void dsfence() {
  asm volatile("s_wait_dscnt 0" ::: "memory");
}

__device__ __forceinline__ float fast_rsq(float x) {
#if __has_builtin(__builtin_amdgcn_rsqf)
  return __builtin_amdgcn_rsqf(x);   // v_rsq_f32, arg >= 1e-5 so no fixups
#else
  return rsqrtf(x);
#endif
}

__device__ __forceinline__ float fast_tanh(float x) {
#if __has_builtin(__builtin_amdgcn_tanhf)
  return __builtin_amdgcn_tanhf(x);  // gfx1250 v_tanh_f32
#else
  float e = __expf(2.f * x);
  return 1.f - 2.f / (e + 1.f);
#endif
}

// Lane-local LayerNorm over an 8-float row (xa=cols0..3, xb=cols4..7).
// Emits the lane's two A-operands (cols {kb,kb+1} and {4+kb,4+kb+1}) with
// gamma/beta already applied (gam4/bet4 hold this lane's 4 column params).
__device__ __forceinline__ void ln_rowA(v4f xa, v4f xb, int h, v4f gam4,
                                        v4f bet4, v2f& A0, v2f& A1) {
  float s = ((xa[0] + xa[1]) + (xa[2] + xa[3])) +
            ((xb[0] + xb[1]) + (xb[2] + xb[3]));
  float q = xa[0] * xa[0];
  q = fmaf(xa[1], xa[1], q); q = fmaf(xa[2], xa[2], q);
  q = fmaf(xa[3], xa[3], q); q = fmaf(xb[0], xb[0], q);
  q = fmaf(xb[1], xb[1], q); q = fmaf(xb[2], xb[2], q);
  q = fmaf(xb[3], xb[3], q);
  float m  = s * 0.125f;
  float v  = fmaf(q, 0.125f, -m * m);
  float rs = fast_rsq(v + 1e-5f);
  float x0 = h ? xa[2] : xa[0];
  float x1 = h ? xa[3] : xa[1];
  float x2 = h ? xb[2] : xb[0];
  float x3 = h ? xb[3] : xb[1];
  A0[0] = fmaf((x0 - m) * rs, gam4[0], bet4[0]);
  A0[1] = fmaf((x1 - m) * rs, gam4[1], bet4[1]);
  A1[0] = fmaf((x2 - m) * rs, gam4[2], bet4[2]);
  A1[1] = fmaf((x3 - m) * rs, gam4[3], bet4[3]);
}

// y (A-operands) -> y + relu(y@W1^T + b1)@W2^T + b2, result in D layout.
// Residual y rides the chain via identity B-operands (augmented K).
__device__ __forceinline__ v8f mlp_path(v2f Ay0, v2f Ay1, const v2f* BW1,
                                        const v2f* BW2, const v2f* BI, v8f Cb1,
                                        v8f Cb2, float* T, int h, int n) {
  const int kb = 2 * h;
  v8f H = wmma4(Ay0, BW1[0], wmma4(Ay1, BW1[1], Cb1));
#pragma unroll
  for (int r = 0; r < 8; ++r) H[r] = fmaxf(H[r], 0.f);
  dsfence();  // previous LDS reads done before overwrite
#pragma unroll
  for (int r = 0; r < 8; ++r) T[(r + 8 * h) * LDS_STRIDE + n] = H[r];
  dsfence();
  v2f Ah[4];
#pragma unroll
  for (int j = 0; j < 4; ++j)
    Ah[j] = *(const v2f*)&T[n * LDS_STRIDE + 4 * j + kb];
  v8f O = wmma4(Ah[0], BW2[0],
          wmma4(Ah[1], BW2[1],
          wmma4(Ah[2], BW2[2],
          wmma4(Ah[3], BW2[3], Cb2))));
  O = wmma4(Ay0, BI[0], wmma4(Ay1, BI[1], O));   // + y residual
  return O;
}

// ---- tiny setup kernel: fold attention constants ---------------------------
// ws[0..7]  = A1  = Wo1*(Wv1*c + bv1) + bo1
// ws[8..15] = b2f = Wo2*bv2 + bo2 + c           (coral residual folded in)
// ws[16..79]= M2  = Wo2*Wv2 (8x8 row-major)
__global__ void atc_setup(const float* __restrict__ coral,
                          const float* __restrict__ wv1, const float* __restrict__ bv1,
                          const float* __restrict__ wo1, const float* __restrict__ bo1,
                          const float* __restrict__ wv2, const float* __restrict__ bv2,
                          const float* __restrict__ wo2, const float* __restrict__ bo2,
                          float* __restrict__ ws) {
  int t = threadIdx.x;
  if (t < 64) {
    int nn = t >> 3, j = t & 7;
    float m = 0.f;
#pragma unroll
    for (int k = 0; k < 8; ++k) m += wo2[nn * 8 + k] * wv2[k * 8 + j];
    ws[16 + nn * 8 + j] = m;
  }
  if (t < 8) {
    float a = bo1[t];
#pragma unroll
    for (int i = 0; i < 8; ++i) {
      float v1i = bv1[i];
#pragma unroll
      for (int j = 0; j < 8; ++j) v1i += wv1[i * 8 + j] * coral[j];
      a += wo1[t * 8 + i] * v1i;
    }
    ws[t] = a;
    float b = bo2[t] + coral[t];
#pragma unroll
    for (int i = 0; i < 8; ++i) b += wo2[t * 8 + i] * bv2[i];
    ws[8 + t] = b;
  }
}

// ---- main fused kernel -----------------------------------------------------
__global__ void __launch_bounds__(256) atc_main(
    const float* __restrict__ g, const float* __restrict__ ws,
    const float* __restrict__ f1b1, const float* __restrict__ f1b2,
    const float* __restrict__ f1w1, const float* __restrict__ f1w2,
    const float* __restrict__ f2b1, const float* __restrict__ f2b2,
    const float* __restrict__ f2w1, const float* __restrict__ f2w2,
    const float* __restrict__ intb, const float* __restrict__ intw,
    const float* __restrict__ ln1b, const float* __restrict__ ln1g,
    const float* __restrict__ ln2b, const float* __restrict__ ln2g,
    float* __restrict__ out, int nRows) {
  __shared__ float lds[8][16 * LDS_STRIDE];   // per-wave transpose tile
  const int lane = threadIdx.x & 31;
  const int wid  = threadIdx.x >> 5;
  float* T = lds[wid];
  const int h  = lane >> 4;     // lane half (0/1)
  const int n  = lane & 15;     // column index (B/C/D) or row index (A)
  const int kb = h << 1;

  // ---- per-wave constant operands (loaded once) ----
  v2f B_I[2], B_M2[2], B_W11[2], B_W12[2], B_W21[4], B_W22[4], B_INT[4];
#pragma unroll
  for (int j = 0; j < 2; ++j)
#pragma unroll
    for (int r = 0; r < 2; ++r) {
      int k = 4 * j + r + kb;                      // B element (k, n)
      B_I[j][r]   = (k == n) ? 1.f : 0.f;          // identity (residual)
      B_M2[j][r]  = (n < 8) ? ws[16 + n * 8 + k] : 0.f;
      B_W11[j][r] = f1w1[n * 8 + k];               // w1 (16,8): B[k][n]=w1[n][k]
      B_W12[j][r] = f2w1[n * 8 + k];
    }
#pragma unroll
  for (int j = 0; j < 4; ++j)
#pragma unroll
    for (int r = 0; r < 2; ++r) {
      int k = 4 * j + r + kb;
      B_W21[j][r] = (n < 8) ? f1w2[n * 16 + k] : 0.f;  // w2 (8,16)
      B_W22[j][r] = (n < 8) ? f2w2[n * 16 + k] : 0.f;
      B_INT[j][r] = (n < 8) ? intw[n * 16 + k] : 0.f;  // int_w (8,16)
    }
  const v8f C_B2F = bcast8((n < 8) ? ws[8 + n] : 0.f);
  const v8f C_B11 = bcast8(f1b1[n]);
  const v8f C_B12 = bcast8(f2b1[n]);
  const v8f C_B21 = bcast8((n < 8) ? f1b2[n] : 0.f);
  const v8f C_B22 = bcast8((n < 8) ? f2b2[n] : 0.f);
  const v8f C_IB  = bcast8((n < 8) ? intb[n] : 0.f);
  // A1 constant (same for all lanes) and per-lane LN column params
  v4f A1lo, A1hi, G1, Be1, G2, Be2;
#pragma unroll
  for (int i = 0; i < 4; ++i) {
    A1lo[i] = ws[i];
    A1hi[i] = ws[4 + i];
    int c = (i < 2) ? (kb + i) : (2 + kb + i);     // cols kb,kb+1,4+kb,5+kb
    G1[i] = ln1g[c];  Be1[i] = ln1b[c];
    G2[i] = ln2g[c];  Be2[i] = ln2b[c];
  }

  const long NK = (long)nRows * 8;
  float* uni = out;
  float* gao = out + NK;
  float* cao = out + 2 * NK;
  float* gwo = out + 3 * NK;
  float* cwo = gwo + nRows;

  const int tiles      = nRows >> 4;
  const int wavesTotal = (gridDim.x * blockDim.x) >> 5;
  const int wave       = (blockIdx.x * blockDim.x + threadIdx.x) >> 5;

  for (int t = wave; t < tiles; t += wavesTotal) {
    const int row0 = t << 4;
    const float* grow = g + (long)(row0 + n) * 8;   // lane n owns row n
    v4f ga4 = *(const v4f*)grow;
    v4f gb4 = *(const v4f*)(grow + 4);

    int tn = t + wavesTotal;                        // prefetch next tile
    if (tn < tiles)
      __builtin_prefetch(g + (long)tn * 128 + (lane & 15) * 8, 0, 3);

    // A-operands of g for the X2 matvec (select lane's 4 columns)
    v2f Ag0, Ag1;
    Ag0[0] = h ? ga4[2] : ga4[0];
    Ag0[1] = h ? ga4[3] : ga4[1];
    Ag1[0] = h ? gb4[2] : gb4[0];
    Ag1[1] = h ? gb4[3] : gb4[1];

    // X2 = g @ M2^T + b2f (D layout) -> LDS -> rows per lane
    v8f X2 = wmma4(Ag0, B_M2[0], wmma4(Ag1, B_M2[1], C_B2F));
    dsfence();
#pragma unroll
    for (int r = 0; r < 8; ++r) T[(r + 8 * h) * LDS_STRIDE + n] = X2[r];
    dsfence();
    v4f x2a = *(const v4f*)&T[n * LDS_STRIDE];
    v4f x2b = *(const v4f*)&T[n * LDS_STRIDE + 4];

    // Path 1: x1 = g + A1, entirely lane-local (no LDS, no WMMA)
    v4f x1a, x1b;
#pragma unroll
    for (int i = 0; i < 4; ++i) { x1a[i] = ga4[i] + A1lo[i]; x1b[i] = gb4[i] + A1hi[i]; }

    v2f Ay10, Ay11, Ay20, Ay21;
    ln_rowA(x1a, x1b, h, G1, Be1, Ay10, Ay11);
    ln_rowA(x2a, x2b, h, G2, Be2, Ay20, Ay21);

    v8f GA = mlp_path(Ay10, Ay11, B_W11, B_W21, B_I, C_B11, C_B21, T, h, n);
    v8f CA = mlp_path(Ay20, Ay21, B_W12, B_W22, B_I, C_B12, C_B22, T, h, n);

    // concat [ga|ca] into LDS (ga incl. zero cols 8..15, ca overwrites them)
    dsfence();
#pragma unroll
    for (int r = 0; r < 8; ++r) T[(r + 8 * h) * LDS_STRIDE + n] = GA[r];
    if (n < 8) {
#pragma unroll
      for (int r = 0; r < 8; ++r) T[(r + 8 * h) * LDS_STRIDE + 8 + n] = CA[r];
    }
    dsfence();
    v2f Ac[4];
#pragma unroll
    for (int j = 0; j < 4; ++j)
      Ac[j] = *(const v2f*)&T[n * LDS_STRIDE + 4 * j + kb];

    v8f U = wmma4(Ac[0], B_INT[0],
            wmma4(Ac[1], B_INT[1],
            wmma4(Ac[2], B_INT[2],
            wmma4(Ac[3], B_INT[3], C_IB))));
#pragma unroll
    for (int r = 0; r < 8; ++r) U[r] = fast_tanh(U[r]);

    // coalesced b128 stores of ga/ca straight from the concat LDS tile
    {
      v4f vga = *(const v4f*)&T[n * LDS_STRIDE + 4 * h];
      v4f vca = *(const v4f*)&T[n * LDS_STRIDE + 8 + 4 * h];
      *(v4f*)(gao + (long)(row0 + n) * 8 + 4 * h) = vga;
      *(v4f*)(cao + (long)(row0 + n) * 8 + 4 * h) = vca;
    }
    dsfence();
#pragma unroll
    for (int r = 0; r < 8; ++r) T[(r + 8 * h) * LDS_STRIDE + n] = U[r];
    dsfence();
    {
      v4f vu = *(const v4f*)&T[n * LDS_STRIDE + 4 * h];
      *(v4f*)(uni + (long)(row0 + n) * 8 + 4 * h) = vu;
    }
    if (lane < 16) {  // attention weights: softmax over 1 key == 1.0
      gwo[row0 + lane] = 1.f;
      cwo[row0 + lane] = 1.f;
    }
  }
}

// ---------------------------------------------------------------------------
// Input order: setup_inputs() dict order, params flattened in jax pytree
// order (nested dicts sorted alphabetically):
//  0 gpcm_betas  1 coral_taus  2 theta
//  3-10  attn1: bk,bo,bq,bv,wk,wo,wq,wv
// 11-18  attn2: bk,bo,bq,bv,wk,wo,wq,wv
// 19-22  ffn1:  b1,b2,w1,w2      23-26  ffn2: b1,b2,w1,w2
// 27 int_b  28 int_w  29 ln1_b  30 ln1_g  31 ln2_b  32 ln2_g
// Output: [unified | ga | ca | gw | cw] flat f32.
// ---------------------------------------------------------------------------
extern "C" void kernel_launch(void* const* d_in, const int* in_sizes, int n_in,
                              void* d_out, int out_size, void* d_ws,
                              size_t ws_size, hipStream_t stream) {
  if (n_in < 33) return;
  const float* g     = (const float*)d_in[0];
  const float* coral = (const float*)d_in[1];
  const float* a1_bo = (const float*)d_in[4];
  const float* a1_bv = (const float*)d_in[6];
  const float* a1_wo = (const float*)d_in[8];
  const float* a1_wv = (const float*)d_in[10];
  const float* a2_bo = (const float*)d_in[12];
  const float* a2_bv = (const float*)d_in[14];
  const float* a2_wo = (const float*)d_in[16];
  const float* a2_wv = (const float*)d_in[18];
  const float* f1b1  = (const float*)d_in[19];
  const float* f1b2  = (const float*)d_in[20];
  const float* f1w1  = (const float*)d_in[21];
  const float* f1w2  = (const float*)d_in[22];
  const float* f2b1  = (const float*)d_in[23];
  const float* f2b2  = (const float*)d_in[24];
  const float* f2w1  = (const float*)d_in[25];
  const float* f2w2  = (const float*)d_in[26];
  const float* intb  = (const float*)d_in[27];
  const float* intw  = (const float*)d_in[28];
  const float* ln1b  = (const float*)d_in[29];
  const float* ln1g  = (const float*)d_in[30];
  const float* ln2b  = (const float*)d_in[31];
  const float* ln2g  = (const float*)d_in[32];
  float* ws  = (float*)d_ws;
  float* out = (float*)d_out;
  const int nRows = in_sizes[0] / 8;   // B*S = 524288

  atc_setup<<<1, 64, 0, stream>>>(coral, a1_wv, a1_bv, a1_wo, a1_bo,
                                  a2_wv, a2_bv, a2_wo, a2_bo, ws);
  // 1024 blocks x 8 waves = 8192 waves; 32768 tiles -> 4 tiles/wave.
  atc_main<<<1024, 256, 0, stream>>>(g, ws, f1b1, f1b2, f1w1, f1w2, f2b1, f2b2,
                                     f2w1, f2w2, intb, intw, ln1b, ln1g, ln2b,
                                     ln2g, out, nRows);
}